// GraphConvolution_28544352649656
// MI455X (gfx1250) — compile-verified
//
#include <hip/hip_runtime.h>
#include <hip/hip_bf16.h>

#define DIN  256
#define DOUT 256

typedef float v2f __attribute__((ext_vector_type(2)));
typedef float v8f __attribute__((ext_vector_type(8)));

// ---------------------------------------------------------------------------
// Kernel 0: out[n, d] = bias[d]   (accumulation target for the SpMM atomics)
// ---------------------------------------------------------------------------
__global__ __launch_bounds__(256) void gcn_init_bias(
    float* __restrict__ out, const float* __restrict__ bias, long total4) {
  long i = (long)blockIdx.x * blockDim.x + threadIdx.x;  // over N*64 float4s
  if (i >= total4) return;
  const float4* b4 = (const float4*)bias;
  ((float4*)out)[i] = b4[i & 63];  // DOUT/4 == 64
}

// ---------------------------------------------------------------------------
// Kernel 1a: weightT[c, k] = weight[k, c]  (256x256, one-time, 256 KB)
// Makes every WMMA B-operand load an 8-byte contiguous global_load_b64.
// ---------------------------------------------------------------------------
__global__ __launch_bounds__(256) void gcn_transpose_w(
    const float* __restrict__ weight, float* __restrict__ weightT) {
  int idx = blockIdx.x * 256 + threadIdx.x;   // grid = 256 blocks
  int k = idx >> 8;                           // row of weight (coalesced read)
  int c = idx & 255;
  weightT[(long)c * DIN + k] = weight[idx];
}

// ---------------------------------------------------------------------------
// Kernel 1b: support = input @ weight  via V_WMMA_F32_16X16X4_F32.
// Register blocking: one wave computes a 32x64 output block
//   (2 M-tiles x 4 N-tiles, 8 v8f accumulators), K advances 4 per WMMA.
//
// Per ISA 7.12.2 f32 operand layouts:
//   A (16x4): lanes 0-15 -> row M=lane, K={k,k+1}; lanes 16-31 -> row M=lane-16,
//             K={k+2,k+3}  (one b64 load per tile per step)
//   B (4x16): mirrored, col N owned per lane; sourced from weightT so the two
//             K values are contiguous (one b64 load per tile per step)
//   C (16x16, 8 VGPRs): VGPR r, lanes 0-15 -> (M=r, N=lane),
//                                lanes 16-31 -> (M=r+8, N=lane-16)
// ---------------------------------------------------------------------------
__global__ __launch_bounds__(256) void gcn_gemm_wmma(
    const float* __restrict__ input,    // [N, DIN]
    const float* __restrict__ weightT,  // [DOUT, DIN] (transposed)
    float* __restrict__ support,        // [N, DOUT]
    int n_rows) {
  const int lane   = threadIdx.x & 31;
  const int waveId = threadIdx.x >> 5;
  const long wave  = (long)blockIdx.x * 8 + waveId;

  const int nGroups = DOUT / 64;                 // 4
  const long mGroup = wave / nGroups;            // 32-row block
  const int  nGroup = (int)(wave % nGroups);     // 64-col block
  const int  rowBase = (int)(mGroup * 32);
  const int  tileN0  = nGroup * 64;
  if (rowBase >= n_rows) return;                 // no barriers -> safe exit

  const int half = lane >> 4;                    // K sub-pair selector
  const int l16  = lane & 15;

  const float* arow0 = input + (long)(rowBase + l16) * DIN;
  const float* arow1 = input + (long)(rowBase + 16 + l16) * DIN;
  const float* brow0 = weightT + (long)(tileN0 + l16) * DIN;       // N-tile 0
  // N-tiles 1..3 are +16 columns each -> +16*DIN floats

  v8f c00 = {}, c01 = {}, c02 = {}, c03 = {};
  v8f c10 = {}, c11 = {}, c12 = {}, c13 = {};

  #pragma unroll 2
  for (int k = 0; k < DIN; k += 4) {
    const int kb = k + 2 * half;
    v2f a0 = *(const v2f*)(arow0 + kb);
    v2f a1 = *(const v2f*)(arow1 + kb);
    v2f b0 = *(const v2f*)(brow0 + kb);
    v2f b1 = *(const v2f*)(brow0 + 16 * DIN + kb);
    v2f b2 = *(const v2f*)(brow0 + 32 * DIN + kb);
    v2f b3 = *(const v2f*)(brow0 + 48 * DIN + kb);
    c00 = __builtin_amdgcn_wmma_f32_16x16x4_f32(false, a0, false, b0, (short)0, c00, false, false);
    c01 = __builtin_amdgcn_wmma_f32_16x16x4_f32(false, a0, false, b1, (short)0, c01, false, false);
    c02 = __builtin_amdgcn_wmma_f32_16x16x4_f32(false, a0, false, b2, (short)0, c02, false, false);
    c03 = __builtin_amdgcn_wmma_f32_16x16x4_f32(false, a0, false, b3, (short)0, c03, false, false);
    c10 = __builtin_amdgcn_wmma_f32_16x16x4_f32(false, a1, false, b0, (short)0, c10, false, false);
    c11 = __builtin_amdgcn_wmma_f32_16x16x4_f32(false, a1, false, b1, (short)0, c11, false, false);
    c12 = __builtin_amdgcn_wmma_f32_16x16x4_f32(false, a1, false, b2, (short)0, c12, false, false);
    c13 = __builtin_amdgcn_wmma_f32_16x16x4_f32(false, a1, false, b3, (short)0, c13, false, false);
  }

  // Store: tile (mt, nt), VGPR r, lane -> row rowBase+mt*16+r+8*half,
  //                                      col tileN0+nt*16+l16
  const v8f acc[2][4] = {{c00, c01, c02, c03}, {c10, c11, c12, c13}};
  #pragma unroll
  for (int mt = 0; mt < 2; ++mt) {
    #pragma unroll
    for (int nt = 0; nt < 4; ++nt) {
      float* dst = support + (long)(rowBase + mt * 16 + 8 * half) * DOUT
                 + tileN0 + nt * 16 + l16;
      #pragma unroll
      for (int r = 0; r < 8; ++r) dst[(long)r * DOUT] = acc[mt][nt][r];
    }
  }
}

// ---------------------------------------------------------------------------
// Kernel 2: out[row[e]] += edge_weight[e] * support[col[e]]
// 64 threads per edge; each thread owns one float4 (16B) of the 256-wide row.
// FP32 hardware atomics (out fits in 192MB L2 -> atomics resolve on-chip).
// ---------------------------------------------------------------------------
__global__ __launch_bounds__(256) void gcn_spmm_scatter(
    const float* __restrict__ support,        // [N, DOUT]
    const long long* __restrict__ edge_index, // [2, E]
    const float* __restrict__ edge_weight,    // [E]
    float* __restrict__ out,                  // [N, DOUT]
    long n_edges) {
  long gid = (long)blockIdx.x * blockDim.x + threadIdx.x;
  long e = gid >> 6;                   // 64 threads per edge
  if (e >= n_edges) return;
  int  j = (int)(gid & 63);            // float4 index within the row

  long r    = edge_index[e];           // destination row
  long cidx = edge_index[n_edges + e]; // source row (gathered)
  float w   = edge_weight[e];

  const float4 v = ((const float4*)(support + cidx * DOUT))[j];
  float* dst = out + r * DOUT + j * 4;
  unsafeAtomicAdd(dst + 0, v.x * w);
  unsafeAtomicAdd(dst + 1, v.y * w);
  unsafeAtomicAdd(dst + 2, v.z * w);
  unsafeAtomicAdd(dst + 3, v.w * w);
}

// ---------------------------------------------------------------------------
extern "C" void kernel_launch(void* const* d_in, const int* in_sizes, int n_in,
                              void* d_out, int out_size, void* d_ws, size_t ws_size,
                              hipStream_t stream) {
  const float*     input       = (const float*)d_in[0];
  const long long* edge_index  = (const long long*)d_in[1];
  const float*     edge_weight = (const float*)d_in[2];
  const float*     weight      = (const float*)d_in[3];
  const float*     bias        = (const float*)d_in[4];
  float*           out         = (float*)d_out;

  const long N = in_sizes[0] / DIN;
  const long E = in_sizes[2];

  float* support = (float*)d_ws;                  // [N, DOUT]  (102.4 MB)
  float* weightT = support + N * (long)DOUT;      // [DOUT, DIN] (256 KB)

  // 1) out = broadcast(bias)
  {
    long total4 = N * (DOUT / 4);
    int  blocks = (int)((total4 + 255) / 256);
    gcn_init_bias<<<blocks, 256, 0, stream>>>(out, bias, total4);
  }
  // 2a) weightT = weight^T
  gcn_transpose_w<<<(DIN * DOUT) / 256, 256, 0, stream>>>(weight, weightT);
  // 2b) support = input @ weight (WMMA f32, 32x64 per wave)
  {
    long waves  = ((N + 31) / 32) * (DOUT / 64);  // 3125 * 4 = 12500
    int  blocks = (int)((waves + 7) / 8);
    gcn_gemm_wmma<<<blocks, 256, 0, stream>>>(input, weightT, support, (int)N);
  }
  // 3) scatter-add over edges
  {
    long threads = E * 64;
    int  blocks  = (int)((threads + 255) / 256);
    gcn_spmm_scatter<<<blocks, 256, 0, stream>>>(support, edge_index,
                                                 edge_weight, out, E);
  }
}